// YoloLayer_27504970564293
// MI455X (gfx1250) — compile-verified
//
#include <hip/hip_runtime.h>
#include <cstdint>

// ---- problem constants (baked from reference) ----
#define NUM_CLASSES 80
#define A_ 3
#define H_ 76
#define W_ 76
#define HW_ (H_ * W_)          // 5776
#define B_ 16
#define N_ (A_ * HW_)          // 17328
#define CH_ (5 + NUM_CLASSES)  // 85 channels per anchor
#define TILE_N 128             // hw positions per tile (tail = 16, still /4)
#define PITCH 132              // floats per LDS row: 528B, 16B aligned, bank-stride 4
#define NTHREADS 256

typedef int   v4i __attribute__((ext_vector_type(4)));
typedef float v4f __attribute__((ext_vector_type(4)));
typedef __attribute__((address_space(1))) v4i* global_v4i_ptr;
typedef __attribute__((address_space(3))) v4i* lds_v4i_ptr;

#if __has_builtin(__builtin_amdgcn_global_load_async_to_lds_b128)
#define HAS_ASYNC_LDS 1
#else
#define HAS_ASYNC_LDS 0
#endif

__device__ __forceinline__ float fsigmoid(float x) {
    float e = __expf(-x);  // v_exp_f32
#if __has_builtin(__builtin_amdgcn_rcpf)
    return __builtin_amdgcn_rcpf(1.0f + e);
#else
    return 1.0f / (1.0f + e);
#endif
}

__device__ __forceinline__ void async_copy16(const float* g, float* l) {
#if HAS_ASYNC_LDS
    __builtin_amdgcn_global_load_async_to_lds_b128(
        (global_v4i_ptr)(uintptr_t)g,   // AS1 v4i32*
        (lds_v4i_ptr)l,                 // AS3 v4i32*
        /*offset=*/0, /*cpol=*/0);
#else
    *(v4f*)l = __builtin_nontemporal_load((const v4f*)g);
#endif
}

__device__ __forceinline__ void wait_async_copies() {
#if HAS_ASYNC_LDS
#if __has_builtin(__builtin_amdgcn_s_wait_asynccnt)
    __builtin_amdgcn_s_wait_asynccnt(0);
#else
    asm volatile("s_wait_asynccnt 0x0" ::: "memory");
#endif
#endif
}

// ---------------------------------------------------------------------------
// Kernel 1: boxes.  One thread per (b, a, hw). Reads channels 0..3 (each read
// coalesced along hw across lanes), writes contiguous float4 per position.
// ---------------------------------------------------------------------------
__global__ __launch_bounds__(NTHREADS) void yolo_boxes_kernel(
    const float* __restrict__ in, float* __restrict__ boxes) {
    int tid = blockIdx.x * NTHREADS + threadIdx.x;
    const int total = B_ * A_ * HW_;
    if (tid >= total) return;

    int hw = tid % HW_;
    int pa = tid / HW_;  // b*A + a
    int a  = pa % A_;
    int b  = pa / A_;

    const float* p = in + (size_t)pa * CH_ * HW_ + hw;
    float x0 = __builtin_nontemporal_load(p + 0 * (size_t)HW_);
    float x1 = __builtin_nontemporal_load(p + 1 * (size_t)HW_);
    float x2 = __builtin_nontemporal_load(p + 2 * (size_t)HW_);
    float x3 = __builtin_nontemporal_load(p + 3 * (size_t)HW_);

    int w = hw % W_;
    int h = hw / W_;

    // masked anchors / stride: (1.5,2.0) (2.375,4.5) (5.0,3.5)
    float aw = (a == 0) ? 1.5f : ((a == 1) ? 2.375f : 5.0f);
    float ah = (a == 0) ? 2.0f : ((a == 1) ? 4.5f   : 3.5f);

    float bx = (fsigmoid(x0) + (float)w) * (1.0f / (float)W_);
    float by = (fsigmoid(x1) + (float)h) * (1.0f / (float)H_);
    float bw = __expf(x2) * (aw * (1.0f / (float)W_));
    float bh = __expf(x3) * (ah * (1.0f / (float)H_));

    float bx1 = bx - 0.5f * bw;
    float by1 = by - 0.5f * bh;
    v4f out4;
    out4.x = bx1;
    out4.y = by1;
    out4.z = bx1 + bw;
    out4.w = by1 + bh;

    size_t n = (size_t)a * HW_ + hw;
    __builtin_nontemporal_store(out4, (v4f*)(boxes + ((size_t)b * N_ + n) * 4));
}

// ---------------------------------------------------------------------------
// Kernel 2: confs.  Tile transpose through LDS using async global->LDS copies.
// One block handles TILE_N consecutive hw positions of one (b,a) plane:
//   stage rows [det, cls0..cls79] x TILE_N into LDS (coalesced B128 async),
//   s_wait_asynccnt + barrier, sigmoid det in place, then emit the fully
//   contiguous [TILE_N x 80] output region with NT stores.
// ---------------------------------------------------------------------------
__global__ __launch_bounds__(NTHREADS) void yolo_confs_kernel(
    const float* __restrict__ in, float* __restrict__ confs) {
    __shared__ float lds[(NUM_CLASSES + 1) * PITCH];  // row 0 = det, rows 1..80 = classes

    const int tilesPerPlane = (HW_ + TILE_N - 1) / TILE_N;  // 46
    int blk   = blockIdx.x;
    int tile  = blk % tilesPerPlane;
    int plane = blk / tilesPerPlane;  // b*A + a
    int a = plane % A_;
    int b = plane / A_;
    int hw0   = tile * TILE_N;
    int valid = min(TILE_N, HW_ - hw0);  // 128 or 16 (both multiples of 4)
    int t = threadIdx.x;

    // base of channel 4 (det) of this plane at column hw0
    const float* src = in + ((size_t)plane * CH_ + 4) * HW_ + hw0;

    // ---- stage: 81 rows x (valid) floats, 16B granules, async to LDS ----
    int granPerRow = valid >> 2;                 // 32 or 4 (power of two)
    int lg         = 31 - __builtin_clz(granPerRow);
    int totalGran  = (NUM_CLASSES + 1) * granPerRow;
    for (int g = t; g < totalGran; g += NTHREADS) {
        int r = g >> lg;
        int j = g & (granPerRow - 1);
        async_copy16(src + (size_t)r * HW_ + (j << 2), &lds[r * PITCH + (j << 2)]);
    }
    wait_async_copies();
    __syncthreads();

    // ---- det sigmoid, computed once per position ----
    if (t < valid) lds[t] = fsigmoid(lds[t]);
    __syncthreads();

    // ---- emit: contiguous [valid x 80] block, NT stores ----
    float* dst = confs + ((size_t)b * N_ + (size_t)a * HW_ + hw0) * NUM_CLASSES;
    int limit = valid * NUM_CLASSES;
    for (int idx = t; idx < limit; idx += NTHREADS) {
        int i = idx / NUM_CLASSES;          // const-80 div -> mul/shift
        int c = idx - i * NUM_CLASSES;
        float v = fsigmoid(lds[(c + 1) * PITCH + i]) * lds[i];
        __builtin_nontemporal_store(v, dst + idx);
    }
}

// ---------------------------------------------------------------------------
extern "C" void kernel_launch(void* const* d_in, const int* in_sizes, int n_in,
                              void* d_out, int out_size, void* d_ws, size_t ws_size,
                              hipStream_t stream) {
    (void)in_sizes; (void)n_in; (void)out_size; (void)d_ws; (void)ws_size;
    const float* in = (const float*)d_in[0];
    float* boxes = (float*)d_out;                          // B*N*4 floats
    float* confs = (float*)d_out + (size_t)B_ * N_ * 4;    // B*N*80 floats

    int boxThreads = B_ * A_ * HW_;  // 277,248
    yolo_boxes_kernel<<<(boxThreads + NTHREADS - 1) / NTHREADS, NTHREADS, 0, stream>>>(in, boxes);

    int tilesPerPlane = (HW_ + TILE_N - 1) / TILE_N;  // 46
    yolo_confs_kernel<<<B_ * A_ * tilesPerPlane, NTHREADS, 0, stream>>>(in, confs);
}